// PyTorchConv1dAutograd_6184752906806
// MI455X (gfx1250) — compile-verified
//
#include <hip/hip_runtime.h>

typedef __attribute__((ext_vector_type(2))) float v2f;
typedef __attribute__((ext_vector_type(8))) float v8f;
typedef __attribute__((ext_vector_type(4))) unsigned int v4u;
typedef __attribute__((ext_vector_type(8))) int v8i;
typedef __attribute__((ext_vector_type(4))) int v4i;

#define N_BATCH 64
#define C_CH    64
#define W_IN    4096
#define F_OUT   64
#define WW      9
#define OUT_W   (W_IN - WW + 1)   // 4088
#define COLS    128               // output columns per workgroup
#define KPC     12                // K per channel, zero-padded from 9 (3 k-steps of 4)
#define FROW    770               // padded filter row stride in floats (even -> 8B aligned, bank-rotating)
#define XROW    144               // x row stride per channel = TDM tile_dim0 (128 + halo + pad)

// Dynamic LDS layout: filt_s[F_OUT*FROW] floats, then x_s[C_CH*XROW] floats.
__global__ __launch_bounds__(256)
void conv1d_wmma_f32(const float* __restrict__ x,
                     const float* __restrict__ filt,
                     const float* __restrict__ bias,
                     float* __restrict__ out)
{
    extern __shared__ float smem[];
    float* filt_s = smem;                 // [f][c*12+w], row stride FROW
    float* x_s    = smem + F_OUT * FROW;  // [c][i], row stride XROW

    const int tid  = threadIdx.x;
    const int lane = tid & 31;
    const int wave = tid >> 5;
    const int n    = blockIdx.y;
    const int t0   = blockIdx.x * COLS;

#if __has_builtin(__builtin_amdgcn_tensor_load_to_lds)
    // ---- TDM: one descriptor DMAs the whole x slab [64 ch][144 dwords] into LDS.
    // tensor_dim0 = W_IN - t0 makes the right-edge halo read zeros (TDM OOB rule).
    if (tid < 32) {
        const unsigned long long ga =
            (unsigned long long)(const void*)(x + (size_t)n * C_CH * W_IN + t0);
        const unsigned int d0 = (unsigned int)(W_IN - t0);

        v4u g0;
        g0.x = 1u;                                               // count=1, user descriptor
        g0.y = (unsigned int)(F_OUT * FROW * sizeof(float));     // lds_addr = x_s byte offset
        g0.z = (unsigned int)(ga & 0xFFFFFFFFu);                 // global_addr[31:0]
        g0.w = (unsigned int)((ga >> 32) & 0x1FFFFFFu) | (2u << 30); // addr[56:32] | type=2

        v8i g1;
        g1[0] = 0x20000;                              // workgroup_mask=0, data_size=2 (4B)
        g1[1] = (int)((d0 & 0xFFFFu) << 16);          // tensor_dim0[15:0] @ bits[63:48]
        g1[2] = (int)((d0 >> 16) | (C_CH << 16));     // tensor_dim0[31:16] | tensor_dim1[15:0]
        g1[3] = (int)(XROW << 16);                    // tensor_dim1[31:16]=0 | tile_dim0=144
        g1[4] = C_CH;                                 // tile_dim1=64 | tile_dim2=0
        g1[5] = W_IN;                                 // tensor_dim0_stride[31:0] = 4096
        g1[6] = 0;                                    // stride0[47:32] | stride1[15:0]
        g1[7] = 0;                                    // stride1[47:16]
        v4i gz = {0, 0, 0, 0};                        // groups 2/3 unused (<=2D tensor)
#if defined(__clang_major__) && (__clang_major__ >= 23)
        v8i gz8 = {0, 0, 0, 0, 0, 0, 0, 0};
        __builtin_amdgcn_tensor_load_to_lds(g0, g1, gz, gz, gz8, 0);
#else
        __builtin_amdgcn_tensor_load_to_lds(g0, g1, gz, gz, 0);
#endif
    }
#endif

    // ---- stage zero-padded filter (overlaps the in-flight TDM) ----
    for (int i = tid; i < F_OUT * C_CH * KPC; i += 256) {
        const int f   = i / (C_CH * KPC);
        const int rem = i - f * (C_CH * KPC);
        const int c   = rem / KPC;
        const int w   = rem - c * KPC;
        float v = 0.0f;
        if (w < WW) v = filt[(f * C_CH + c) * WW + w];
        filt_s[f * FROW + rem] = v;
    }

#if __has_builtin(__builtin_amdgcn_tensor_load_to_lds)
    if (tid < 32) __builtin_amdgcn_s_wait_tensorcnt(0);
#else
    // Fallback: plain staged copy of the x slab.
    const float* xn = x + (size_t)n * C_CH * W_IN;
    for (int j = tid; j < C_CH * XROW; j += 256) {
        const int c = j / XROW;
        const int i = j - c * XROW;
        const int g = t0 + i;
        float v = 0.0f;
        if (i < COLS + (WW - 1) && g < W_IN) v = xn[c * W_IN + g];
        x_s[j] = v;
    }
#endif
    __syncthreads();

    const int strip = wave * 16;                 // this wave's 16-column strip
    const int lm    = lane & 15;                 // M (A) / N (B,D) index within tile
    const int khalf = (lane < 16) ? 0 : 2;       // K offset selected by lane half

    v8f acc0 = {}; v8f acc1 = {}; v8f acc2 = {}; v8f acc3 = {};

    for (int c = 0; c < C_CH; ++c) {
        const float* xc = &x_s[c * XROW + strip + lm];
        const float* fc = &filt_s[c * KPC + lm * FROW];
#pragma unroll
        for (int ks = 0; ks < 3; ++ks) {
            const int w0 = ks * 4 + khalf;
            // B fragment (4x16): lane holds column N=lm, K = w0(,+1); B[k][t] = x[c, t + k]
            v2f b;
            b.x = xc[w0];
            b.y = xc[w0 + 1];
            // A fragments (16x4) for the 4 F-tiles: lane holds row M=lm, K = w0(,+1)
            const float* fl = fc + w0;
            v2f a0, a1, a2, a3;
            a0.x = fl[0 * 16 * FROW]; a0.y = fl[0 * 16 * FROW + 1];
            a1.x = fl[1 * 16 * FROW]; a1.y = fl[1 * 16 * FROW + 1];
            a2.x = fl[2 * 16 * FROW]; a2.y = fl[2 * 16 * FROW + 1];
            a3.x = fl[3 * 16 * FROW]; a3.y = fl[3 * 16 * FROW + 1];
            acc0 = __builtin_amdgcn_wmma_f32_16x16x4_f32(false, a0, false, b, (short)0, acc0, false, false);
            acc1 = __builtin_amdgcn_wmma_f32_16x16x4_f32(false, a1, false, b, (short)0, acc1, false, false);
            acc2 = __builtin_amdgcn_wmma_f32_16x16x4_f32(false, a2, false, b, (short)0, acc2, false, false);
            acc3 = __builtin_amdgcn_wmma_f32_16x16x4_f32(false, a3, false, b, (short)0, acc3, false, false);
        }
    }

    // ---- store D + C_CH * bias[f]; D layout: VGPR r -> M = r (lanes 0-15) / 8+r (lanes 16-31), N = lm ----
    const int t_out = t0 + strip + lm;
    if (t_out < OUT_W) {
        float* outn = out + (size_t)n * F_OUT * OUT_W;
#pragma unroll
        for (int r = 0; r < 8; ++r) {
            const int frow = (lane < 16) ? r : (8 + r);
            outn[(size_t)(frow +  0) * OUT_W + t_out] = acc0[r] + (float)C_CH * bias[frow +  0];
            outn[(size_t)(frow + 16) * OUT_W + t_out] = acc1[r] + (float)C_CH * bias[frow + 16];
            outn[(size_t)(frow + 32) * OUT_W + t_out] = acc2[r] + (float)C_CH * bias[frow + 32];
            outn[(size_t)(frow + 48) * OUT_W + t_out] = acc3[r] + (float)C_CH * bias[frow + 48];
        }
    }
}

extern "C" void kernel_launch(void* const* d_in, const int* in_sizes, int n_in,
                              void* d_out, int out_size, void* d_ws, size_t ws_size,
                              hipStream_t stream) {
    const float* x    = (const float*)d_in[0];
    const float* filt = (const float*)d_in[1];
    const float* bias = (const float*)d_in[2];
    float* out        = (float*)d_out;

    dim3 grid(W_IN / COLS, N_BATCH);  // 32 column-blocks x 64 batches
    const size_t shmem = (size_t)(F_OUT * FROW + C_CH * XROW) * sizeof(float); // ~229 KB (<=320 KB/WGP)
    conv1d_wmma_f32<<<grid, dim3(256), shmem, stream>>>(x, filt, bias, out);
}